// Encoder_NRI_24275155157246
// MI455X (gfx1250) — compile-verified
//
#include <hip/hip_runtime.h>

// CDNA5 / gfx1250, wave32. fp32 WMMA: V_WMMA_F32_16X16X4_F32.
typedef __attribute__((ext_vector_type(2))) float v2f;
typedef __attribute__((ext_vector_type(8))) float v8f;

__device__ __forceinline__ v8f wmma_f32(v2f a, v2f b, v8f c) {
  // 8 args: (neg_a, A, neg_b, B, c_mod, C, reuse_a, reuse_b)
  return __builtin_amdgcn_wmma_f32_16x16x4_f32(false, a, false, b, (short)0, c,
                                               false, false);
}

// ---------------------------------------------------------------------------
// K1: h = relu(x @ Wn1 + bn1) @ Wn2 + bn2        x:(1024,6) -> h:(1024,128)
// ---------------------------------------------------------------------------
__global__ __launch_bounds__(256) void k_node_ffn(
    const float* __restrict__ x, const float* __restrict__ Wn1,
    const float* __restrict__ bn1, const float* __restrict__ Wn2,
    const float* __restrict__ bn2, float* __restrict__ h) {
  __shared__ float xs[16][8];
  __shared__ float ts[16][256];
  const int tid = threadIdx.x;
  const int node0 = blockIdx.x * 16;

  if (tid < 16 * 8) {
    int r = tid >> 3, k = tid & 7;
    xs[r][k] = (k < 6) ? x[(node0 + r) * 6 + k] : 0.0f;
  }
  __syncthreads();

  {  // layer 1 (K=6): each thread owns one of the 256 output channels
    int o = tid;
    float w0 = Wn1[0 * 256 + o], w1 = Wn1[1 * 256 + o], w2 = Wn1[2 * 256 + o];
    float w3 = Wn1[3 * 256 + o], w4 = Wn1[4 * 256 + o], w5 = Wn1[5 * 256 + o];
    float b = bn1[o];
    for (int r = 0; r < 16; ++r) {
      float acc = b + xs[r][0] * w0 + xs[r][1] * w1 + xs[r][2] * w2 +
                  xs[r][3] * w3 + xs[r][4] * w4 + xs[r][5] * w5;
      ts[r][o] = fmaxf(acc, 0.0f);
    }
  }
  __syncthreads();

  // layer 2: (16x256)@(256x128) via WMMA; wave w owns cols [16w,16w+16)
  const int wave = tid >> 5, lane = tid & 31;
  const int hi = lane >> 4, lc = lane & 15;
  const int n0 = wave * 16;
  v8f acc = {};
  for (int k0 = 0; k0 < 256; k0 += 4) {
    int k = k0 + 2 * hi;
    v2f a, b;
    a.x = ts[lc][k];
    a.y = ts[lc][k + 1];
    b.x = Wn2[k * 128 + n0 + lc];
    b.y = Wn2[(k + 1) * 128 + n0 + lc];
    acc = wmma_f32(a, b, acc);
  }
  float bias = bn2[n0 + lc];
  for (int v = 0; v < 8; ++v) {
    int r = v + 8 * hi;
    h[(node0 + r) * 128 + n0 + lc] = acc[v] + bias;
  }
}

// ---------------------------------------------------------------------------
// K2: E = relu(h @ We1f + be1) @ We2 + be2       (We1 folded: rows c + c+128)
// ---------------------------------------------------------------------------
__global__ __launch_bounds__(256) void k_edge_ffn(
    const float* __restrict__ h, const float* __restrict__ We1,
    const float* __restrict__ be1, const float* __restrict__ We2,
    const float* __restrict__ be2, float* __restrict__ E) {
  __shared__ float hs[16][128];
  __shared__ float us[16][128];
  const int tid = threadIdx.x;
  const int node0 = blockIdx.x * 16;
  for (int idx = tid; idx < 16 * 128; idx += 256)
    hs[idx >> 7][idx & 127] = h[node0 * 128 + idx];
  __syncthreads();

  const int wave = tid >> 5, lane = tid & 31;
  const int hi = lane >> 4, lc = lane & 15;
  const int n0 = wave * 16;

  v8f acc = {};
  for (int k0 = 0; k0 < 128; k0 += 4) {
    int k = k0 + 2 * hi;
    v2f a, b;
    a.x = hs[lc][k];
    a.y = hs[lc][k + 1];
    b.x = We1[k * 128 + n0 + lc] + We1[(k + 128) * 128 + n0 + lc];
    b.y = We1[(k + 1) * 128 + n0 + lc] + We1[(k + 129) * 128 + n0 + lc];
    acc = wmma_f32(a, b, acc);
  }
  {
    float bias = be1[n0 + lc];
    for (int v = 0; v < 8; ++v) us[v + 8 * hi][n0 + lc] = fmaxf(acc[v] + bias, 0.0f);
  }
  __syncthreads();

  v8f acc2 = {};
  for (int k0 = 0; k0 < 128; k0 += 4) {
    int k = k0 + 2 * hi;
    v2f a, b;
    a.x = us[lc][k];
    a.y = us[lc][k + 1];
    b.x = We2[k * 128 + n0 + lc];
    b.y = We2[(k + 1) * 128 + n0 + lc];
    acc2 = wmma_f32(a, b, acc2);
  }
  float bias2 = be2[n0 + lc];
  for (int v = 0; v < 8; ++v) {
    int r = v + 8 * hi;
    E[(node0 + r) * 128 + n0 + lc] = acc2[v] + bias2;
  }
}

// ---------------------------------------------------------------------------
// K3: S[n][c] = sum_i E[n,i,c]   (deterministic, single block)
// ---------------------------------------------------------------------------
__global__ __launch_bounds__(256) void k_reduce(const float* __restrict__ E,
                                                float* __restrict__ S) {
  int tid = threadIdx.x;  // 256 = 2 batches * 128 channels
  int n = tid >> 7, c = tid & 127;
  const float* base = E + (size_t)n * 512 * 128 + c;
  float s = 0.0f;
  for (int i = 0; i < 512; ++i) s += base[i * 128];
  S[n * 128 + c] = s;
}

// ---------------------------------------------------------------------------
// K4: h2 = relu((S - E) @ Wen1 + ben1) @ Wen2 + ben2  -> d_out[0 : 1024*128)
// ---------------------------------------------------------------------------
__global__ __launch_bounds__(256) void k_e2n_ffn(
    const float* __restrict__ E, const float* __restrict__ S,
    const float* __restrict__ Wen1, const float* __restrict__ ben1,
    const float* __restrict__ Wen2, const float* __restrict__ ben2,
    float* __restrict__ h2) {
  __shared__ float as[16][128];
  __shared__ float us[16][256];
  const int tid = threadIdx.x;
  const int node0 = blockIdx.x * 16;
  for (int idx = tid; idx < 16 * 128; idx += 256) {
    int r = idx >> 7, c = idx & 127;
    int node = node0 + r;
    as[r][c] = S[(node >> 9) * 128 + c] - E[node * 128 + c];
  }
  __syncthreads();

  const int wave = tid >> 5, lane = tid & 31;
  const int hi = lane >> 4, lc = lane & 15;

  // GEMM1: (16x128)@(128x256): 16 col-tiles, 2 per wave
  for (int t = 0; t < 2; ++t) {
    int n0 = (wave + t * 8) * 16;
    v8f acc = {};
    for (int k0 = 0; k0 < 128; k0 += 4) {
      int k = k0 + 2 * hi;
      v2f a, b;
      a.x = as[lc][k];
      a.y = as[lc][k + 1];
      b.x = Wen1[k * 256 + n0 + lc];
      b.y = Wen1[(k + 1) * 256 + n0 + lc];
      acc = wmma_f32(a, b, acc);
    }
    float bias = ben1[n0 + lc];
    for (int v = 0; v < 8; ++v) us[v + 8 * hi][n0 + lc] = fmaxf(acc[v] + bias, 0.0f);
  }
  __syncthreads();

  // GEMM2: (16x256)@(256x128)
  int n0 = wave * 16;
  v8f acc2 = {};
  for (int k0 = 0; k0 < 256; k0 += 4) {
    int k = k0 + 2 * hi;
    v2f a, b;
    a.x = us[lc][k];
    a.y = us[lc][k + 1];
    b.x = Wen2[k * 128 + n0 + lc];
    b.y = Wen2[(k + 1) * 128 + n0 + lc];
    acc2 = wmma_f32(a, b, acc2);
  }
  float bias2 = ben2[n0 + lc];
  for (int v = 0; v < 8; ++v) {
    int r = v + 8 * hi;
    h2[(node0 + r) * 128 + n0 + lc] = acc2[v] + bias2;
  }
}

// ---------------------------------------------------------------------------
// K5: scalar[node] = (relu(h2 @ Wne1f + bne1) . Wne2 + bne2) / 512
//     out2[node, j] = scalar[node]  for all j in [0,512)
// ---------------------------------------------------------------------------
__global__ __launch_bounds__(256) void k_n2e(
    const float* __restrict__ h2, const float* __restrict__ Wne1,
    const float* __restrict__ bne1, const float* __restrict__ Wne2,
    const float* __restrict__ bne2, float* __restrict__ out2) {
  __shared__ float hs[16][128];
  __shared__ float us[16][128];
  __shared__ float ps[16][16];
  __shared__ float sc[16];
  const int tid = threadIdx.x;
  const int node0 = blockIdx.x * 16;
  for (int idx = tid; idx < 16 * 128; idx += 256)
    hs[idx >> 7][idx & 127] = h2[node0 * 128 + idx];
  __syncthreads();

  const int wave = tid >> 5, lane = tid & 31;
  const int hi = lane >> 4, lc = lane & 15;
  const int n0 = wave * 16;
  v8f acc = {};
  for (int k0 = 0; k0 < 128; k0 += 4) {
    int k = k0 + 2 * hi;
    v2f a, b;
    a.x = hs[lc][k];
    a.y = hs[lc][k + 1];
    b.x = Wne1[k * 128 + n0 + lc] + Wne1[(k + 128) * 128 + n0 + lc];
    b.y = Wne1[(k + 1) * 128 + n0 + lc] + Wne1[(k + 129) * 128 + n0 + lc];
    acc = wmma_f32(a, b, acc);
  }
  float bias = bne1[n0 + lc];
  for (int v = 0; v < 8; ++v) us[v + 8 * hi][n0 + lc] = fmaxf(acc[v] + bias, 0.0f);
  __syncthreads();

  {  // dot with Wne2 (128 -> 1): 16 partial lanes per row
    int r = tid >> 4, l16 = tid & 15;
    float p = 0.0f;
    for (int c = l16; c < 128; c += 16) p += us[r][c] * Wne2[c];
    ps[r][l16] = p;
  }
  __syncthreads();
  if (tid < 16) {
    float s = 0.0f;
    for (int i = 0; i < 16; ++i) s += ps[tid][i];
    sc[tid] = (s + bne2[0]) * (1.0f / 512.0f);
  }
  __syncthreads();

  for (int idx = tid; idx < 16 * 512; idx += 256) {
    int r = idx >> 9, j = idx & 511;
    out2[(size_t)(node0 + r) * 512 + j] = sc[r];
  }
}

// ---------------------------------------------------------------------------
extern "C" void kernel_launch(void* const* d_in, const int* in_sizes, int n_in,
                              void* d_out, int out_size, void* d_ws,
                              size_t ws_size, hipStream_t stream) {
  (void)in_sizes; (void)n_in; (void)out_size; (void)ws_size;
  const float* x    = (const float*)d_in[0];
  const float* Wn1  = (const float*)d_in[1];
  const float* bn1  = (const float*)d_in[2];
  const float* Wn2  = (const float*)d_in[3];
  const float* bn2  = (const float*)d_in[4];
  const float* We1  = (const float*)d_in[5];
  const float* be1  = (const float*)d_in[6];
  const float* We2  = (const float*)d_in[7];
  const float* be2  = (const float*)d_in[8];
  const float* Wen1 = (const float*)d_in[9];
  const float* ben1 = (const float*)d_in[10];
  const float* Wen2 = (const float*)d_in[11];
  const float* ben2 = (const float*)d_in[12];
  const float* Wne1 = (const float*)d_in[13];
  const float* bne1 = (const float*)d_in[14];
  const float* Wne2 = (const float*)d_in[15];
  const float* bne2 = (const float*)d_in[16];

  float* ws = (float*)d_ws;
  float* h  = ws;                    // 1024*128
  float* E  = ws + 1024 * 128;       // 1024*128
  float* S  = ws + 2 * 1024 * 128;   // 2*128
  float* h2   = (float*)d_out;                // outputs: h2 then edge2
  float* out2 = (float*)d_out + 1024 * 128;

  dim3 blk(256);
  k_node_ffn<<<64, blk, 0, stream>>>(x, Wn1, bn1, Wn2, bn2, h);
  k_edge_ffn<<<64, blk, 0, stream>>>(h, We1, be1, We2, be2, E);
  k_reduce<<<1, blk, 0, stream>>>(E, S);
  k_e2n_ffn<<<64, blk, 0, stream>>>(E, S, Wen1, ben1, Wen2, ben2, h2);
  k_n2e<<<64, blk, 0, stream>>>(h2, Wne1, bne1, Wne2, bne2, out2);
}